// ChamferDistancePadL2_5248450036648
// MI455X (gfx1250) — compile-verified
//
#include <hip/hip_runtime.h>

typedef __attribute__((ext_vector_type(2))) float v2f;
typedef __attribute__((ext_vector_type(8))) float v8f;

__device__ __forceinline__ float lane_shfl(float v, int src) {
    return __shfl(v, src, 32);
}
__device__ __forceinline__ float lane_shfl_xor(float v, int mask) {
    return __shfl_xor(v, mask, 32);
}

// ---------------------------------------------------------------------------
// Precompute the WMMA-ready "B-form" of a cloud (used as the column matrix).
// For each 16-point tile t, 32 float2 slots laid out so that wave lane L
// loads slot (t*32 + L) as one coalesced b64 and is immediately WMMA-ready:
//   slot s  (s<16) : (-2*y0, -2*y1)  of point t*16+s      -> B rows K=0,1
//   slot s (s>=16) : (-2*y2, |y|^2)  of point t*16+s-16   -> B rows K=2,3
// ---------------------------------------------------------------------------
__global__ __launch_bounds__(256) void prep_form_kernel(
    const float* __restrict__ pts, float2* __restrict__ form, int total)
{
    int p = blockIdx.x * blockDim.x + threadIdx.x;
    if (p >= total) return;
    const float* s = pts + (size_t)p * 3;
    float y0 = s[0], y1 = s[1], y2 = s[2];
    float sq = fmaf(y2, y2, fmaf(y1, y1, y0 * y0));
    int tile = p >> 4, k = p & 15;
    form[(size_t)tile * 32 + k]      = make_float2(-2.0f * y0, -2.0f * y1);
    form[(size_t)tile * 32 + 16 + k] = make_float2(-2.0f * y2, sq);
}

// ---------------------------------------------------------------------------
// One direction of Chamfer. Each wave owns TWO 16-row tiles of P (32 rows)
// and sweeps all M columns, 32 per iteration (2 precomputed B tiles).
// Loop body: 2 coalesced b64 loads -> 4 x V_WMMA_F32_16X16X4_F32 -> 16 min3.
//
//   A row i  = [x0, x1, x2, 1]
//   B col j  = [-2y0, -2y1, -2y2, |y|^2]   (precomputed)
//   C[i][j]  = |x_i|^2                     (row bias via lane shuffles)
//   D = A*B + C = full squared distance; running min in accumulator VGPRs.
// ---------------------------------------------------------------------------
__global__ __launch_bounds__(256) void chamfer_dir_kernel(
    const float* __restrict__ P,       // [B, N, 3] raw rows cloud
    const float2* __restrict__ Qform,  // [B, M/16, 32] precomputed cols form
    float* __restrict__ partials,      // [B * N/32] per-wave sums
    int N, int M)
{
    const int tid           = threadIdx.x;
    const int lane          = tid & 31;
    const int waveInBlock   = tid >> 5;
    const int wavesPerBlock = blockDim.x >> 5;
    const int b             = blockIdx.y;
    const int pairIdx       = blockIdx.x * wavesPerBlock + waveInBlock;
    const int pairsPerBatch = gridDim.x * wavesPerBlock;   // == N/32
    const int n0            = pairIdx * 32;

    const int  sub = lane & 15;
    const bool hi  = lane >= 16;

    // ---- Build two A tiles (rows n0..n0+15 and n0+16..n0+31).
    const float* pb0 = P + ((size_t)b * N + (size_t)(n0 + sub)) * 3;
    const float* pb1 = pb0 + 48;  // +16 points * 3
    float x0 = pb0[0], x1 = pb0[1], x2 = pb0[2];
    float w0 = pb1[0], w1 = pb1[1], w2 = pb1[2];
    float sqx0 = fmaf(x2, x2, fmaf(x1, x1, x0 * x0));
    float sqx1 = fmaf(w2, w2, fmaf(w1, w1, w0 * w0));
    v2f a0, a1;
    a0.x = hi ? x2 : x0;  a0.y = hi ? 1.0f : x1;
    a1.x = hi ? w2 : w0;  a1.y = hi ? 1.0f : w1;

    // ---- Row-norm bias tiles: cbias[r] = |x|^2 of row (r + 8*half).
    const int basesrc = hi ? 8 : 0;
    v8f cb0, cb1;
#pragma unroll
    for (int r = 0; r < 8; ++r) {
        cb0[r] = lane_shfl(sqx0, basesrc + r);
        cb1[r] = lane_shfl(sqx1, basesrc + r);
    }

    v8f mn0, mn1;
#pragma unroll
    for (int r = 0; r < 8; ++r) { mn0[r] = 3.0e38f; mn1[r] = 3.0e38f; }

    // ---- Sweep columns: 32 per iteration, 4 WMMAs.
    const float2* qf = Qform + (size_t)b * (M / 16) * 32 + lane;
    const int nTiles = M / 16;
    for (int t = 0; t < nTiles; t += 2) {
        v2f bv0 = *(const v2f*)&qf[(size_t)t * 32];
        v2f bv1 = *(const v2f*)&qf[(size_t)(t + 1) * 32];

        v8f d00 = __builtin_amdgcn_wmma_f32_16x16x4_f32(
            false, a0, false, bv0, (short)0, cb0, false, false);
        v8f d01 = __builtin_amdgcn_wmma_f32_16x16x4_f32(
            false, a0, false, bv1, (short)0, cb0, false, false);
        v8f d10 = __builtin_amdgcn_wmma_f32_16x16x4_f32(
            false, a1, false, bv0, (short)0, cb1, false, false);
        v8f d11 = __builtin_amdgcn_wmma_f32_16x16x4_f32(
            false, a1, false, bv1, (short)0, cb1, false, false);

#pragma unroll
        for (int r = 0; r < 8; ++r) {
            mn0[r] = fminf(mn0[r], fminf(d00[r], d01[r]));  // -> v_min3_num_f32
            mn1[r] = fminf(mn1[r], fminf(d10[r], d11[r]));
        }
    }

    // ---- Min across the 16 column-lanes (masks stay within each half).
#pragma unroll
    for (int mask = 1; mask <= 8; mask <<= 1) {
#pragma unroll
        for (int r = 0; r < 8; ++r) {
            mn0[r] = fminf(mn0[r], lane_shfl_xor(mn0[r], mask));
            mn1[r] = fminf(mn1[r], lane_shfl_xor(mn1[r], mask));
        }
    }

    // ---- Clamp (max(.,0) commutes with min) and sum this wave's 32 rows.
    float rowsum = 0.0f;
#pragma unroll
    for (int r = 0; r < 8; ++r)
        rowsum += fmaxf(mn0[r], 0.0f) + fmaxf(mn1[r], 0.0f);
    float tot = rowsum + lane_shfl_xor(rowsum, 16);  // combine lane halves

    if (lane == 0) partials[(size_t)b * pairsPerBatch + pairIdx] = tot;
}

// Deterministic fixed-shape reduction: sum1*scale1 + sum2*scale2 -> out[0].
__global__ __launch_bounds__(256) void chamfer_reduce_kernel(
    const float* __restrict__ partials, float* __restrict__ out,
    int n1, int n2, float scale1, float scale2)
{
    __shared__ float s1[256];
    __shared__ float s2[256];
    const int t = threadIdx.x;
    float a = 0.0f, b = 0.0f;
    for (int i = t; i < n1; i += 256) a += partials[i];
    for (int i = t; i < n2; i += 256) b += partials[n1 + i];
    s1[t] = a; s2[t] = b;
    __syncthreads();
    for (int s = 128; s > 0; s >>= 1) {
        if (t < s) { s1[t] += s1[t + s]; s2[t] += s2[t + s]; }
        __syncthreads();
    }
    if (t == 0) out[0] = s1[0] * scale1 + s2[0] * scale2;
}

extern "C" void kernel_launch(void* const* d_in, const int* in_sizes, int n_in,
                              void* d_out, int out_size, void* d_ws, size_t ws_size,
                              hipStream_t stream) {
    const float* xyz1 = (const float*)d_in[0];
    const float* xyz2 = (const float*)d_in[1];
    float* out = (float*)d_out;

    const int B = 16;
    const int N = in_sizes[0] / (B * 3);   // 4096
    const int M = in_sizes[1] / (B * 3);   // 4096

    // Workspace layout: [form1: B*N*2 float2][form2: B*M*2 float2][partials]
    char* ws = (char*)d_ws;
    float2* form1 = (float2*)ws;                               // xyz1 as cols
    float2* form2 = (float2*)(ws + (size_t)B * N * 2 * 8);     // xyz2 as cols
    float*  partials = (float*)(ws + (size_t)B * N * 2 * 8 + (size_t)B * M * 2 * 8);

    const int t1 = B * (N / 32);   // 2048 waves, dir 1
    const int t2 = B * (M / 32);   // 2048 waves, dir 2

    // Precompute WMMA-ready column forms of both clouds.
    {
        int total1 = B * N, total2 = B * M;
        prep_form_kernel<<<(total1 + 255) / 256, 256, 0, stream>>>(xyz1, form1, total1);
        prep_form_kernel<<<(total2 + 255) / 256, 256, 0, stream>>>(xyz2, form2, total2);
    }

    const int wavesPerBlock = 8;   // 256 threads = 8 wave32
    dim3 block(256);
    dim3 grid1((N / 32) / wavesPerBlock, B);   // 16 x 16 blocks
    dim3 grid2((M / 32) / wavesPerBlock, B);

    // dist1: rows = xyz1, cols = xyz2 (form2)
    chamfer_dir_kernel<<<grid1, block, 0, stream>>>(xyz1, form2, partials, N, M);
    // dist2: rows = xyz2, cols = xyz1 (form1)
    chamfer_dir_kernel<<<grid2, block, 0, stream>>>(xyz2, form1, partials + t1, M, N);

    chamfer_reduce_kernel<<<1, 256, 0, stream>>>(
        partials, out, t1, t2, 1.0f / (float)(B * N), 1.0f / (float)(B * M));
}